// NeighborListWithCutoff_6657199309587
// MI455X (gfx1250) — compile-verified
//
#include <hip/hip_runtime.h>

typedef __attribute__((ext_vector_type(2))) float v2f;
typedef __attribute__((ext_vector_type(4))) float v4f;
typedef __attribute__((ext_vector_type(8))) float v8f;

#define CUTOFF_NM 1.0f

// ---------------------------------------------------------------------------
// Pass 0: per-atom squared norm |x|^2 into workspace (32 KB, stays L2-hot).
// ---------------------------------------------------------------------------
__global__ void nl_sqnorm_kernel(const float* __restrict__ coords,
                                 float* __restrict__ sq, int n) {
    int i = blockIdx.x * blockDim.x + threadIdx.x;
    if (i < n) {
        float x = coords[3 * i + 0];
        float y = coords[3 * i + 1];
        float z = coords[3 * i + 2];
        sq[i] = x * x + y * y + z * z;
    }
}

// ---------------------------------------------------------------------------
// Pass 1: ii / jj index planes. Pure index function -> fully coalesced
// b128 non-temporal streaming stores (512 MiB of the 1 GiB output).
// ---------------------------------------------------------------------------
__global__ void nl_index_planes_kernel(float* __restrict__ out, int n) {
    unsigned k = (blockIdx.x * blockDim.x + threadIdx.x) * 4u;  // N^2 < 2^32
    unsigned un = (unsigned)n;
    unsigned row = k / un;
    unsigned col = k - row * un;
    size_t nn = (size_t)n * (size_t)n;

    float fr = (float)row;
    v4f ii = {fr, fr, fr, fr};
    v4f jj = {(float)col, (float)(col + 1), (float)(col + 2), (float)(col + 3)};
    __builtin_nontemporal_store(ii, (v4f*)(out + k));
    __builtin_nontemporal_store(jj, (v4f*)(out + nn + k));
}

// ---------------------------------------------------------------------------
// Pass 2: one wave per 16x16 pair tile.
//   gram tile   : V_WMMA_F32_16X16X4_F32  (K=3 padded to 4 with zero)
//   A layout    : lanes 0-15 hold (x,y) of row M=lane, lanes 16-31 hold (z,0)
//   B layout    : same striping for columns
//   C/D layout  : VGPR r, lane half h -> (row = i0+8h+r, col = j0+(lane&15))
// Epilogue: r2 = sq_i + sq_j - 2*gram, dist, molecule/cutoff mask; NT stores.
// Block = 256 threads = 8 waves = 8 row-tiles x 1 col-tile (128x16 region).
// ---------------------------------------------------------------------------
__global__ void nl_tiles_kernel(const float* __restrict__ coords,
                                const int* __restrict__ mol,
                                const float* __restrict__ sq,
                                float* __restrict__ out, int n) {
    const int lane = threadIdx.x & 31;
    const int wave = threadIdx.x >> 5;
    const int half = lane >> 4;   // 0: K=0,1 lanes   1: K=2,3 lanes
    const int l16  = lane & 15;

    const int i0 = (blockIdx.y * 8 + wave) * 16;  // row tile base
    const int j0 = blockIdx.x * 16;               // col tile base

    // A (16x4, rows i0..i0+15) and B (4x16, cols j0..j0+15), K padded to 4.
    int arow = i0 + l16;
    int bcol = j0 + l16;
    v2f a, b;
    if (half == 0) {
        a.x = coords[3 * arow + 0];
        a.y = coords[3 * arow + 1];
        b.x = coords[3 * bcol + 0];
        b.y = coords[3 * bcol + 1];
    } else {
        a.x = coords[3 * arow + 2];
        a.y = 0.0f;
        b.x = coords[3 * bcol + 2];
        b.y = 0.0f;
    }

    v8f c = {};
    // 8 args: (neg_a, A, neg_b, B, c_mod, C, reuse_a, reuse_b)
    c = __builtin_amdgcn_wmma_f32_16x16x4_f32(
        false, a, false, b, (short)0, c, false, false);

    // Column-side scalars for this lane (L2/L0 broadcast hits).
    float sqj = sq[j0 + l16];
    int   mj  = mol[j0 + l16];

    // Row-side scalars: 8 rows covered by this lane's accumulator VGPRs.
    const int rbase = i0 + 8 * half;
    float sqi[8];
    int   mi[8];
#pragma unroll
    for (int r = 0; r < 8; ++r) {
        sqi[r] = sq[rbase + r];
        mi[r]  = mol[rbase + r];
    }

    size_t nn = (size_t)n * (size_t)n;
    float* __restrict__ distp = out + 2 * nn;
    float* __restrict__ maskp = out + 3 * nn;
    const int col = j0 + l16;

#pragma unroll
    for (int r = 0; r < 8; ++r) {
        int row = rbase + r;
        float r2 = sqi[r] + sqj - 2.0f * c[r];
        float d  = sqrtf(fmaxf(r2, 0.0f));
        bool within = (mi[r] == mj) && (row != col) && (d <= CUTOFF_NM);
        size_t off = (size_t)row * (size_t)n + (size_t)col;
        __builtin_nontemporal_store(within ? d : 0.0f, distp + off);
        __builtin_nontemporal_store(within ? 1.0f : 0.0f, maskp + off);
    }
}

// ---------------------------------------------------------------------------
// Inputs (setup_inputs order):
//   d_in[0] coordinates            float32 (N,3)   in_sizes[0] = 3N
//   d_in[1] atomic_subsystem_ids   int32   (N,)    in_sizes[1] = N
// Output: 4 flat float planes [ii | jj | distances | mask], each N*N.
// Workspace: sq[] = N floats at d_ws offset 0.
// ---------------------------------------------------------------------------
extern "C" void kernel_launch(void* const* d_in, const int* in_sizes, int n_in,
                              void* d_out, int out_size, void* d_ws, size_t ws_size,
                              hipStream_t stream) {
    const float* coords = (const float*)d_in[0];
    const int*   mol    = (const int*)d_in[1];
    const int n = in_sizes[1];          // 8192
    float* out = (float*)d_out;
    float* sq  = (float*)d_ws;

    nl_sqnorm_kernel<<<(n + 255) / 256, 256, 0, stream>>>(coords, sq, n);

    size_t nn = (size_t)n * (size_t)n;
    unsigned idx_blocks = (unsigned)(nn / 4 / 256);   // 65536 for N=8192
    nl_index_planes_kernel<<<idx_blocks, 256, 0, stream>>>(out, n);

    dim3 grid(n / 16, n / 128);         // 512 col tiles x 64 row groups
    nl_tiles_kernel<<<grid, 256, 0, stream>>>(coords, mol, sq, out, n);
}